// SAM_75325136437769
// MI455X (gfx1250) — compile-verified
//
#include <hip/hip_runtime.h>

// SAM dynamic local convolution, MI455X/gfx1250, fp32 WMMA pipeline.
//  K0: concat w1|w2|w3 -> Wc[320,256], b1|b2|b3 -> bc[320]   (one-shot, tiny)
//  K1: fused conv1x1 for x1|x2|x3  (O=320,K=256 GEMM, 4 o-tiles/wave)
//  K2: reflect-pad unfold of x2 + BN1 + ReLU -> Hin[B,320,HW]
//  K3: cw1 GEMM (O=32,K=320, 2 o-tiles/wave) + BN2 + ReLU -> Hg[B,32,HW]
//  K4: cw2 GEMM (O=288,K=32, 3 o-tiles/wave) + bias       -> Wk[B,288,HW]
//  K5: grouped 3x3 local conv (zero pad)  -> out[B,256,H,W]
// o-dimension register blocking shares each 4x16 activation (B) fragment
// across OT WMMAs; weight sub-tiles are addressed off ONE base pointer with
// compile-time offsets so the loads stay on the GLOBAL path (global_load_b64
// with immediate offsets), not FLAT.

#define HWD   4096       // 64*64
#define CINC  256
#define BATCH 8
#define MIDC  320
#define SAM_EPS 1e-5f

typedef __attribute__((ext_vector_type(2))) float v2f;
typedef __attribute__((ext_vector_type(8))) float v8f;

// OT 16x16 o-subtiles x one 16-wide p-tile, K in steps of 4 via
// V_WMMA_F32_16X16X4_F32.  Fragment layout per CDNA5 ISA 7.12.2:
//   lane=(h<<4)|j : A vgpr v <- W[row j, k0+2h+v]   (wq pre-offset, +t*TSTRIDE)
//                   B vgpr v <- X[k0+2h+v, col j]   (xp pre-offset)
//   D   vgpr v -> element (m=8h+v, n=j)
template <int OT, int TSTRIDE>   // TSTRIDE = 16*ldw (elements)
__device__ __forceinline__ void wmma_gemm_multi(const float* __restrict__ wq,
                                                const float* __restrict__ xp,
                                                int ldx, int ksteps,
                                                v8f (&acc)[OT]) {
  #pragma unroll 2
  for (int k = 0; k < ksteps; ++k) {
    v2f bf;
    bf[0] = xp[0];
    bf[1] = xp[ldx];
    #pragma unroll
    for (int t = 0; t < OT; ++t) {
      v2f af;
      af[0] = wq[t * TSTRIDE + 0];
      af[1] = wq[t * TSTRIDE + 1];
      acc[t] = __builtin_amdgcn_wmma_f32_16x16x4_f32(false, af, false, bf,
                                                     (short)0, acc[t],
                                                     false, false);
    }
    wq += 4;
    xp += (size_t)4 * ldx;
  }
}

// K0: Wc[o,c] = concat(w1,w2,w3); bc[o] = concat(b1,b2,b3).  320*256 threads.
__global__ void __launch_bounds__(256)
k0_concat_w(const float* __restrict__ w1, const float* __restrict__ b1,
            const float* __restrict__ w2, const float* __restrict__ b2,
            const float* __restrict__ w3, const float* __restrict__ b3,
            float* __restrict__ Wc, float* __restrict__ bc) {
  const int i = blockIdx.x * 256 + threadIdx.x;     // [0, 320*256)
  float wv;
  if (i < 32 * CINC)      wv = w1[i];
  else if (i < 64 * CINC) wv = w2[i - 32 * CINC];
  else                    wv = w3[i - 64 * CINC];
  Wc[i] = wv;
  if (i < MIDC)
    bc[i] = (i < 32) ? b1[i] : (i < 64) ? b2[i - 32] : b3[i - 64];
}

// K1: A[b,o,p] = sum_c Wc[o,c]*x[b,c,p] + bc[o], o in [0,320).
// grid (32, 5, 8), block 256 (8 waves, one p-tile each, 4 o-tiles per wave).
__global__ void __launch_bounds__(256)
k1_conv1x1_x123(const float* __restrict__ x, const float* __restrict__ Wc,
                const float* __restrict__ bc, float* __restrict__ A) {
  const int lane = threadIdx.x & 31;
  const int wave = threadIdx.x >> 5;
  const int j = lane & 15, h = lane >> 4;
  const int pBase = (blockIdx.x * 8 + wave) * 16;
  const int oBase0 = blockIdx.y * 64;
  const int b = blockIdx.z;

  v8f acc[4] = {};
  const float* wq = Wc + (size_t)(oBase0 + j) * CINC + 2 * h;
  const float* xp = x + (size_t)b * CINC * HWD + (size_t)(2 * h) * HWD + pBase + j;
  wmma_gemm_multi<4, 16 * CINC>(wq, xp, HWD, CINC / 4, acc);

  #pragma unroll
  for (int t = 0; t < 4; ++t) {
    float* outp = A + (size_t)b * MIDC * HWD + (size_t)(oBase0 + 16 * t) * HWD
                + pBase + j;
    #pragma unroll
    for (int v = 0; v < 8; ++v) {
      const int r = 8 * h + v;
      outp[(size_t)r * HWD] = acc[t][v] + bc[oBase0 + 16 * t + r];
    }
  }
}

// K2: Hin[b,c,p] = relu(bn1(concat(x1, unfold_reflect(x2))[c,p]))
__global__ void __launch_bounds__(256)
k2_unfold_bn(const float* __restrict__ A,
             const float* __restrict__ g1, const float* __restrict__ bb1,
             const float* __restrict__ m1, const float* __restrict__ vv1,
             float* __restrict__ Hin) {
  const int idx = blockIdx.x * 256 + threadIdx.x;   // B*320*HW threads
  const int p = idx & (HWD - 1);
  const int c = (idx >> 12) % MIDC;
  const int b = idx / (MIDC * HWD);

  float val;
  const float* Ab = A + (size_t)b * MIDC * HWD;
  if (c < 32) {
    val = Ab[(size_t)c * HWD + p];                  // x1
  } else {
    const int cc = c - 32;
    const int c2 = cc / 9;
    const int kk = cc % 9;
    int hh = (p >> 6) + (kk / 3) - 1;
    int ww = (p & 63) + (kk % 3) - 1;
    hh = (hh < 0) ? 1 : ((hh > 63) ? 62 : hh);      // reflect pad=1
    ww = (ww < 0) ? 1 : ((ww > 63) ? 62 : ww);
    val = Ab[(size_t)(32 + c2) * HWD + (hh << 6) + ww];   // x2 shifted
  }
  const float s = g1[c] * rsqrtf(vv1[c] + SAM_EPS);
  val = val * s + (bb1[c] - m1[c] * s);
  Hin[idx] = fmaxf(val, 0.0f);
}

// K3: Hg = relu(bn2(cw1 @ Hin)), O=32, K=320.  grid (32, 1, 8).
__global__ void __launch_bounds__(256)
k3_gemm_mid(const float* __restrict__ Hin, const float* __restrict__ cw1,
            const float* __restrict__ g2, const float* __restrict__ bb2,
            const float* __restrict__ m2, const float* __restrict__ vv2,
            float* __restrict__ Hg) {
  const int lane = threadIdx.x & 31;
  const int wave = threadIdx.x >> 5;
  const int j = lane & 15, h = lane >> 4;
  const int pBase = (blockIdx.x * 8 + wave) * 16;
  const int b = blockIdx.z;

  v8f acc[2] = {};
  const float* wq = cw1 + (size_t)j * MIDC + 2 * h;
  const float* xp = Hin + (size_t)b * MIDC * HWD + (size_t)(2 * h) * HWD + pBase + j;
  wmma_gemm_multi<2, 16 * MIDC>(wq, xp, HWD, MIDC / 4, acc);

  #pragma unroll
  for (int t = 0; t < 2; ++t) {
    float* outp = Hg + (size_t)b * 32 * HWD + (size_t)(16 * t) * HWD + pBase + j;
    #pragma unroll
    for (int v = 0; v < 8; ++v) {
      const int c = 16 * t + 8 * h + v;
      const float s = g2[c] * rsqrtf(vv2[c] + SAM_EPS);
      const float val = acc[t][v] * s + (bb2[c] - m2[c] * s);
      outp[(size_t)(8 * h + v) * HWD] = fmaxf(val, 0.0f);
    }
  }
}

// K4: Wk = cw2 @ Hg + cb2, O=288, K=32.  grid (32, 6, 8).
__global__ void __launch_bounds__(256)
k4_gemm_w(const float* __restrict__ Hg, const float* __restrict__ cw2,
          const float* __restrict__ cb2, float* __restrict__ Wk) {
  const int lane = threadIdx.x & 31;
  const int wave = threadIdx.x >> 5;
  const int j = lane & 15, h = lane >> 4;
  const int pBase = (blockIdx.x * 8 + wave) * 16;
  const int oBase0 = blockIdx.y * 48;
  const int b = blockIdx.z;

  v8f acc[3] = {};
  const float* wq = cw2 + (size_t)(oBase0 + j) * 32 + 2 * h;
  const float* xp = Hg + (size_t)b * 32 * HWD + (size_t)(2 * h) * HWD + pBase + j;
  wmma_gemm_multi<3, 16 * 32>(wq, xp, HWD, 32 / 4, acc);

  #pragma unroll
  for (int t = 0; t < 3; ++t) {
    float* outp = Wk + (size_t)b * 288 * HWD + (size_t)(oBase0 + 16 * t) * HWD
                + pBase + j;
    #pragma unroll
    for (int v = 0; v < 8; ++v) {
      const int c = oBase0 + 16 * t + 8 * h + v;
      outp[(size_t)(8 * h + v) * HWD] = acc[t][v] + cb2[c];
    }
  }
}

// K5: out[b,o,p] = sum_kk x3[b,o, p shifted by kk (zero pad)] * Wk[b, g*9+kk, p]
__global__ void __launch_bounds__(256)
k5_localconv(const float* __restrict__ A, const float* __restrict__ Wk,
             float* __restrict__ out) {
  const int idx = blockIdx.x * 256 + threadIdx.x;   // B*256*HW threads
  const int p = idx & (HWD - 1);
  const int o = (idx >> 12) & 255;
  const int b = idx >> 20;
  const int g = o >> 3;
  const int hh = p >> 6, ww = p & 63;

  const float* x3 = A + (size_t)b * MIDC * HWD + (size_t)(64 + o) * HWD;
  const float* wk = Wk + (size_t)b * 288 * HWD + (size_t)g * 9 * HWD + p;

  float acc = 0.0f;
  #pragma unroll
  for (int kk = 0; kk < 9; ++kk) {
    const int hi = hh + kk / 3 - 1;
    const int wi = ww + kk % 3 - 1;
    const bool ok = (hi >= 0) & (hi < 64) & (wi >= 0) & (wi < 64);
    const float xv = ok ? x3[(hi << 6) + wi] : 0.0f;
    acc = fmaf(xv, wk[(size_t)kk * HWD], acc);
  }
  out[idx] = acc;
}

extern "C" void kernel_launch(void* const* d_in, const int* in_sizes, int n_in,
                              void* d_out, int out_size, void* d_ws, size_t ws_size,
                              hipStream_t stream) {
  (void)in_sizes; (void)n_in; (void)out_size; (void)ws_size;
  const float* x     = (const float*)d_in[0];
  const float* w1    = (const float*)d_in[1];
  const float* b1    = (const float*)d_in[2];
  const float* w2    = (const float*)d_in[3];
  const float* b2    = (const float*)d_in[4];
  const float* w3    = (const float*)d_in[5];
  const float* b3    = (const float*)d_in[6];
  const float* bn1_g = (const float*)d_in[7];
  const float* bn1_b = (const float*)d_in[8];
  const float* bn1_m = (const float*)d_in[9];
  const float* bn1_v = (const float*)d_in[10];
  const float* cw1   = (const float*)d_in[11];
  const float* bn2_g = (const float*)d_in[12];
  const float* bn2_b = (const float*)d_in[13];
  const float* bn2_m = (const float*)d_in[14];
  const float* bn2_v = (const float*)d_in[15];
  const float* cw2   = (const float*)d_in[16];
  const float* cb2   = (const float*)d_in[17];

  float* ws = (float*)d_ws;
  const size_t szA  = (size_t)BATCH * MIDC * HWD;     // 10,485,760 floats
  const size_t szHg = (size_t)BATCH * 32 * HWD;       //  1,048,576 floats
  float* A   = ws;                                    // x1|x2|x3, [B,320,HW]
  float* Hin = ws + szA;                              // [B,320,HW]
  float* Hg  = ws + 2 * szA;                          // [B,32,HW]
  float* Wc  = ws + 2 * szA + szHg;                   // [320,256]
  float* bc  = Wc + (size_t)MIDC * CINC;              // [320]
  float* Wk  = Hin;                                   // reuse (Hin dead after K3)

  k0_concat_w<<<(MIDC * CINC) / 256, 256, 0, stream>>>(
      w1, b1, w2, b2, w3, b3, Wc, bc);

  k1_conv1x1_x123<<<dim3(32, 5, BATCH), 256, 0, stream>>>(x, Wc, bc, A);

  k2_unfold_bn<<<(BATCH * MIDC * HWD) / 256, 256, 0, stream>>>(
      A, bn1_g, bn1_b, bn1_m, bn1_v, Hin);

  k3_gemm_mid<<<dim3(32, 1, BATCH), 256, 0, stream>>>(
      Hin, cw1, bn2_g, bn2_b, bn2_m, bn2_v, Hg);

  k4_gemm_w<<<dim3(32, 6, BATCH), 256, 0, stream>>>(Hg, cw2, cb2, Wk);

  k5_localconv<<<(BATCH * 256 * HWD) / 256, 256, 0, stream>>>(
      A, Wk, (float*)d_out);
}